// GCN_molecule_regression_10230612099847
// MI455X (gfx1250) — compile-verified
//
#include <hip/hip_runtime.h>

typedef __attribute__((ext_vector_type(2))) float v2f;
typedef __attribute__((ext_vector_type(8))) float v8f;

#define HDIM 64

// ---------------------------------------------------------------------------
// Generic fill
// ---------------------------------------------------------------------------
__global__ void fill_kernel(float* __restrict__ p, long long n, float v) {
    long long i = blockIdx.x * (long long)blockDim.x + threadIdx.x;
    if (i < n) p[i] = v;
}

// ---------------------------------------------------------------------------
// Degree: deg pre-filled with 1.0 (self loop); accumulate in-degree
// ---------------------------------------------------------------------------
__global__ void deg_accum_kernel(const long long* __restrict__ dst, float* __restrict__ deg, int E) {
    int e = blockIdx.x * blockDim.x + threadIdx.x;
    if (e < E) atomicAdd(&deg[(int)dst[e]], 1.0f);
}

__global__ void dinv_kernel(float* __restrict__ deg, int N) {
    int i = blockIdx.x * blockDim.x + threadIdx.x;
    if (i < N) deg[i] = rsqrtf(deg[i]);   // in place: deg -> dinv
}

// ---------------------------------------------------------------------------
// GEMM: Hout[N x 64] = X[N x K] @ W[K x 64], fp32 WMMA 16x16x4.
// One wave per 16-row M tile; 4 accumulators cover all 64 output columns.
// Launch: grid = ceil(N/16) blocks of 32 threads (1 wave -> EXEC all ones).
// ---------------------------------------------------------------------------
__global__ void gemm_wmma_kernel(const float* __restrict__ X, const float* __restrict__ W,
                                 float* __restrict__ Hout, int K, int N) {
    const int mt   = blockIdx.x;
    const int lane = threadIdx.x;        // blockDim.x == 32
    const int n    = lane & 15;          // column-in-tile / row-in-tile for A
    const int g    = lane >> 4;          // half-wave group
    const int m0   = mt * 16;

    int arow = m0 + n;                   // A row this lane supplies
    if (arow >= N) arow = N - 1;         // clamp (N % 16 == 0 for this problem)
    const float* xr = X + (long long)arow * K;

    v8f acc0 = {}, acc1 = {}, acc2 = {}, acc3 = {};

    for (int k0 = 0; k0 < K; k0 += 4) {
        const int kk = k0 + 2 * g;       // ISA A layout: lanes 0-15 K={k0,k0+1}, 16-31 K={k0+2,k0+3}
        v2f a;  a.x = xr[kk];  a.y = xr[kk + 1];

        const float* wr = W + (long long)kk * HDIM;       // B[kk][*], next row +HDIM
        v2f b0; b0.x = wr[ 0 + n]; b0.y = wr[HDIM +  0 + n];
        v2f b1; b1.x = wr[16 + n]; b1.y = wr[HDIM + 16 + n];
        v2f b2; b2.x = wr[32 + n]; b2.y = wr[HDIM + 32 + n];
        v2f b3; b3.x = wr[48 + n]; b3.y = wr[HDIM + 48 + n];

        acc0 = __builtin_amdgcn_wmma_f32_16x16x4_f32(false, a, false, b0, (short)0, acc0, false, false);
        acc1 = __builtin_amdgcn_wmma_f32_16x16x4_f32(false, a, false, b1, (short)0, acc1, false, false);
        acc2 = __builtin_amdgcn_wmma_f32_16x16x4_f32(false, a, false, b2, (short)0, acc2, false, false);
        acc3 = __builtin_amdgcn_wmma_f32_16x16x4_f32(false, a, false, b3, (short)0, acc3, false, false);
    }

    // C/D layout: VGPR r, half g -> row m0 + r + 8*g, column n (+16*tile)
    for (int r = 0; r < 8; ++r) {
        const int m = m0 + r + 8 * g;
        if (m < N) {
            float* orow = Hout + (long long)m * HDIM + n;
            orow[ 0] = acc0[r];
            orow[16] = acc1[r];
            orow[32] = acc2[r];
            orow[48] = acc3[r];
        }
    }
}

// ---------------------------------------------------------------------------
// Scatter: one wave per (edge | self-loop). y[dst] += h[src] * dinv[s]*dinv[d]
// ---------------------------------------------------------------------------
__global__ void scatter_kernel(const float* __restrict__ h, const long long* __restrict__ src,
                               const long long* __restrict__ dst, const float* __restrict__ dinv,
                               float* __restrict__ y, int E, int N) {
    const long long wid  = (blockIdx.x * (long long)blockDim.x + threadIdx.x) >> 5;
    const int       lane = threadIdx.x & 31;
    const long long total = (long long)E + N;
    if (wid >= total) return;

    int s, d;
    if (wid < E) { s = (int)src[wid]; d = (int)dst[wid]; }
    else         { s = d = (int)(wid - E); }

    const float nrm = dinv[s] * dinv[d];
    const float* hs = h + (long long)s * HDIM;
    float*       yd = y + (long long)d * HDIM;
    atomicAdd(&yd[lane],      hs[lane]      * nrm);
    atomicAdd(&yd[lane + 32], hs[lane + 32] * nrm);
}

// ---------------------------------------------------------------------------
// y = relu(y + b), in place over N*HDIM
// ---------------------------------------------------------------------------
__global__ void bias_relu_kernel(float* __restrict__ y, const float* __restrict__ b, long long n) {
    long long i = blockIdx.x * (long long)blockDim.x + threadIdx.x;
    if (i < n) y[i] = fmaxf(y[i] + b[i & (HDIM - 1)], 0.0f);
}

// ---------------------------------------------------------------------------
// Pooling: one wave per node. Sum + count + max (uint-bit atomicMax; x >= 0).
// ---------------------------------------------------------------------------
__global__ void pool_kernel(const float* __restrict__ xf, const long long* __restrict__ batch,
                            float* __restrict__ psum, float* __restrict__ pmax,
                            float* __restrict__ cnt, int N) {
    const long long wid  = (blockIdx.x * (long long)blockDim.x + threadIdx.x) >> 5;
    const int       lane = threadIdx.x & 31;
    if (wid >= N) return;

    const int g = (int)batch[wid];
    const float* xr = xf + wid * HDIM;
    const float v0 = xr[lane], v1 = xr[lane + 32];

    atomicAdd(&psum[g * HDIM + lane],      v0);
    atomicAdd(&psum[g * HDIM + lane + 32], v1);
    atomicMax((unsigned int*)&pmax[g * HDIM + lane],      __float_as_uint(v0));
    atomicMax((unsigned int*)&pmax[g * HDIM + lane + 32], __float_as_uint(v1));
    if (lane == 0) atomicAdd(&cnt[g], 1.0f);
}

// ---------------------------------------------------------------------------
// Finalize: per graph -> xc=[mean|max] into out[G..], dot(xc,Wout)+bout into out[0..G)
// ---------------------------------------------------------------------------
__global__ void finalize_kernel(const float* __restrict__ psum, const float* __restrict__ pmax,
                                const float* __restrict__ cnt, const float* __restrict__ Wout,
                                const float* __restrict__ bout, float* __restrict__ out, int G) {
    const int g = blockIdx.x;
    const int t = threadIdx.x;                  // 128 threads
    const float c = fmaxf(cnt[g], 1.0f);
    const float v = (t < HDIM) ? (psum[g * HDIM + t] / c) : pmax[g * HDIM + (t - HDIM)];
    out[G + (long long)g * (2 * HDIM) + t] = v;

    __shared__ float red[2 * HDIM];
    red[t] = v * Wout[t];
    __syncthreads();
    for (int s = HDIM; s > 0; s >>= 1) {
        if (t < s) red[t] += red[t + s];
        __syncthreads();
    }
    if (t == 0) out[g] = red[0] + bout[0];
}

// ---------------------------------------------------------------------------
// Host orchestration
// ---------------------------------------------------------------------------
static inline size_t align256(size_t s) { return (s + 255) & ~(size_t)255; }

extern "C" void kernel_launch(void* const* d_in, const int* in_sizes, int n_in,
                              void* d_out, int out_size, void* d_ws, size_t ws_size,
                              hipStream_t stream) {
    const float*      x     = (const float*)d_in[0];
    const long long*  edge  = (const long long*)d_in[1];   // [2, E] int64
    const long long*  batch = (const long long*)d_in[2];   // [N]    int64
    const float* W0 = (const float*)d_in[3];  const float* b0 = (const float*)d_in[4];
    const float* W1 = (const float*)d_in[5];  const float* b1 = (const float*)d_in[6];
    const float* W2 = (const float*)d_in[7];  const float* b2 = (const float*)d_in[8];
    const float* W3 = (const float*)d_in[9];  const float* b3 = (const float*)d_in[10];
    const float* Wout = (const float*)d_in[11];
    const float* bout = (const float*)d_in[12];
    float* out = (float*)d_out;

    const int N  = in_sizes[2];
    const int E  = in_sizes[1] / 2;
    const int K0 = in_sizes[0] / N;                 // 128
    const int G  = out_size / (2 * HDIM + 1);       // 2048

    const long long* src = edge;
    const long long* dst = edge + E;

    // Workspace carve-up
    char* ws = (char*)d_ws;
    float* dinv = (float*)ws;  ws += align256((size_t)N * 4);
    float* hbuf = (float*)ws;  ws += align256((size_t)N * HDIM * 4);
    float* bufA = (float*)ws;  ws += align256((size_t)N * HDIM * 4);
    float* bufB = (float*)ws;  ws += align256((size_t)N * HDIM * 4);
    float* psum = (float*)ws;  ws += align256((size_t)G * HDIM * 4);
    float* pmax = (float*)ws;  ws += align256((size_t)G * HDIM * 4);
    float* cnt  = (float*)ws;  ws += align256((size_t)G * 4);
    (void)ws_size; (void)n_in;

    const long long NH = (long long)N * HDIM;
    const int TPB = 256;
    dim3 blk(TPB);
    auto blocks1d = [](long long n, int t) { return dim3((unsigned)((n + t - 1) / t)); };

    // ---- degree / dinv ----
    fill_kernel<<<blocks1d(N, TPB), blk, 0, stream>>>(dinv, N, 1.0f);  // self loop
    deg_accum_kernel<<<blocks1d(E, TPB), blk, 0, stream>>>(dst, dinv, E);
    dinv_kernel<<<blocks1d(N, TPB), blk, 0, stream>>>(dinv, N);

    // ---- 4 GCN layers ----
    const float* Wl[4] = {W0, W1, W2, W3};
    const float* bl[4] = {b0, b1, b2, b3};
    const float* xin = x;
    const int mtiles = (N + 15) / 16;
    const long long scatter_waves = (long long)E + N;
    dim3 scatter_grid = blocks1d(scatter_waves * 32, TPB);

    for (int l = 0; l < 4; ++l) {
        const int K = (l == 0) ? K0 : HDIM;
        float* y = (l & 1) ? bufB : bufA;

        gemm_wmma_kernel<<<dim3(mtiles), dim3(32), 0, stream>>>(xin, Wl[l], hbuf, K, N);
        fill_kernel<<<blocks1d(NH, TPB), blk, 0, stream>>>(y, NH, 0.0f);
        scatter_kernel<<<scatter_grid, blk, 0, stream>>>(hbuf, src, dst, dinv, y, E, N);
        bias_relu_kernel<<<blocks1d(NH, TPB), blk, 0, stream>>>(y, bl[l], NH);
        xin = y;
    }

    // ---- pooling ----
    fill_kernel<<<blocks1d((long long)G * HDIM, TPB), blk, 0, stream>>>(psum, (long long)G * HDIM, 0.0f);
    fill_kernel<<<blocks1d((long long)G * HDIM, TPB), blk, 0, stream>>>(pmax, (long long)G * HDIM, 0.0f);
    fill_kernel<<<blocks1d(G, TPB), blk, 0, stream>>>(cnt, G, 0.0f);
    pool_kernel<<<blocks1d((long long)N * 32, TPB), blk, 0, stream>>>(xin, batch, psum, pmax, cnt, N);

    // ---- readout ----
    finalize_kernel<<<dim3(G), dim3(2 * HDIM), 0, stream>>>(psum, pmax, cnt, Wout, bout, out, G);
}